// MaxAggregator_9182640078907
// MI455X (gfx1250) — compile-verified
//
#include <hip/hip_runtime.h>
#include <stdint.h>

// Ragged neighbor gather + per-node max reduce for MI455X (gfx1250).
// out[n, d] = max_{s<NUM_S} features[nbr[n,s], d]
//
// features table (102.4 MB) fits in the 192 MB L2 -> the 1.02 GB random
// gather is L2-bound. CDNA5 async-to-LDS (ASYNCcnt) keeps 40 B128 gather
// requests in flight per wave with no VGPR staging, double-buffered in LDS.
// Output stores are non-temporal so they don't evict the feature table
// from L2 (table + output > 192 MB otherwise).

#define NUM_S      10
#define D_FEAT_    256
#define LANES      32
#define WPB        4                    // waves per block (128 threads)
#define NPW        8                    // nodes processed per wave
#define BUF_FLOATS (NUM_S * D_FEAT_)    // 2560 floats = 10 KB per buffer
#define BUF_BYTES  (BUF_FLOATS * 4)

typedef float v4f __attribute__((ext_vector_type(4)));

// Issue 20 async B128 gathers for one node's 10 neighbor rows.
// GVS form: saddr = row base (SGPR pair), vaddr = lane*16 (loop-invariant
// VGPR), vdst = per-buffer LDS base + lane*16 (loop-invariant VGPR).
// The instruction immediate offset is added to BOTH the LDS and the global
// address (CDNA5 ISA), so the per-neighbor LDS stride s*1024
// (+512 for the upper half-row) lives in the immediate and is compensated
// on the global side via saddr = feat + (idx<<10) - s*1024.
__device__ __forceinline__ void async_gather_node(
    const int* __restrict__ nbr, const char* feat_b,
    int node, uint32_t lds_base, uint32_t voff)
{
    int idxs[NUM_S];
#pragma unroll
    for (int s = 0; s < NUM_S; ++s)      // node is wave-uniform -> SMEM loads,
        idxs[s] = nbr[node * NUM_S + s]; // batched, single kmcnt wait
#pragma unroll
    for (int s = 0; s < NUM_S; ++s) {
        const char* row = feat_b
                        + (((uint64_t)(uint32_t)idxs[s]) << 10)  // zero-extend
                        - (uint32_t)(s * 1024);
        asm volatile("global_load_async_to_lds_b128 %0, %1, %2 offset:%3"
                     :: "v"(lds_base), "v"(voff), "s"(row), "n"(s * 1024)
                     : "memory");
        asm volatile("global_load_async_to_lds_b128 %0, %1, %2 offset:%3"
                     :: "v"(lds_base), "v"(voff), "s"(row), "n"(s * 1024 + 512)
                     : "memory");
    }
}

__global__ __launch_bounds__(WPB * LANES)
void maxagg_async_kernel(const int* __restrict__ nbr,
                         const float* __restrict__ feat,
                         float* __restrict__ out,
                         int n_nodes)
{
    __shared__ float lds[WPB * 2 * BUF_FLOATS];   // 80 KB / block
    const int lane = (int)(threadIdx.x & (LANES - 1));
    const int wave = (int)(threadIdx.x >> 5);
    float* wbuf = &lds[wave * 2 * BUF_FLOATS];

    const uint32_t voff = (uint32_t)lane * 16u;                 // vaddr VGPR
    const uint32_t ldsA = (uint32_t)(uintptr_t)wbuf + voff;     // buffer 0
    const uint32_t ldsB = ldsA + BUF_BYTES;                     // buffer 1
    const char* feat_b = (const char*)feat;

    int base = ((int)blockIdx.x * WPB + wave) * NPW;
    base = __builtin_amdgcn_readfirstlane(base);
    int cnt = n_nodes - base;
    if (cnt <= 0) return;                 // no barriers used -> safe early exit
    if (cnt > NPW) cnt = NPW;

    // Prologue: fill buffer 0.
    async_gather_node(nbr, feat_b, base, ldsA, voff);

    for (int k = 0; k < cnt; ++k) {
        float* cur = wbuf + (k & 1) * BUF_FLOATS;
        const bool more = (k + 1) < cnt;
        if (more) {
            // Issue next node into the other buffer (its DS reads completed
            // one iteration ago), then wait for the current buffer: async
            // loads complete in order, so asynccnt <= 20 => buffer k done.
            async_gather_node(nbr, feat_b, base + k + 1,
                              ((k + 1) & 1) ? ldsB : ldsA, voff);
            asm volatile("s_wait_asynccnt %0" :: "n"(2 * NUM_S) : "memory");
        } else {
            asm volatile("s_wait_asynccnt 0" ::: "memory");
        }

        // Consume: 10 rows from LDS (ds_load_b128), componentwise max.
        const v4f* cb = (const v4f*)cur;
        v4f a0 = cb[lane];
        v4f a1 = cb[LANES + lane];
#pragma unroll
        for (int s = 1; s < NUM_S; ++s) {
            const v4f* sb = (const v4f*)(cur + s * D_FEAT_);
            a0 = __builtin_elementwise_max(a0, sb[lane]);
            a1 = __builtin_elementwise_max(a1, sb[LANES + lane]);
        }

        // Coalesced non-temporal B128 stores: wave writes the full 1 KB row
        // without evicting the L2-resident feature table.
        v4f* orow = (v4f*)(out + (size_t)(base + k) * D_FEAT_);
        __builtin_nontemporal_store(a0, orow + lane);
        __builtin_nontemporal_store(a1, orow + LANES + lane);
    }
}

extern "C" void kernel_launch(void* const* d_in, const int* in_sizes, int n_in,
                              void* d_out, int out_size, void* d_ws, size_t ws_size,
                              hipStream_t stream) {
    const int*   nbr  = (const int*)d_in[0];    // [N, 10] int32
    const float* feat = (const float*)d_in[1];  // [N, 256] f32
    float*       out  = (float*)d_out;          // [N, 256] f32

    const int n_nodes = in_sizes[0] / NUM_S;    // 100000
    const int nodes_per_block = WPB * NPW;      // 32
    const int blocks = (n_nodes + nodes_per_block - 1) / nodes_per_block; // 3125

    maxagg_async_kernel<<<blocks, WPB * LANES, 0, stream>>>(nbr, feat, out, n_nodes);
}